// EdgeWeightNetwork_541165879643
// MI455X (gfx1250) — compile-verified
//
#include <hip/hip_runtime.h>
#include <hip/hip_bf16.h>
#include <math.h>

typedef __attribute__((ext_vector_type(2))) float v2f;
typedef __attribute__((ext_vector_type(8))) float v8f;

// ---------------------------------------------------------------------------
// Kernel 1: node projection  proj[n] = { nf[n] . W[0:128],  nf[n] . W[128:256] }
// One wave32 handles a 16-node tile via V_WMMA_F32_16X16X4_F32, K-loop of 32.
// A (16x4 f32): lanes 0-15 -> M=lane, K=k0+{0,1}; lanes 16-31 -> M=lane-16, K=k0+{2,3}
// B (4x16 f32): VGPR0 -> K = half*2,   VGPR1 -> K = half*2+1; N = lane&15.
//               Only N=0 (W[:128]) and N=1 (W[128:]) are nonzero; implemented
//               branch-free as (per-lane column base) + (0/1 float mask).
// D (16x16 f32): VGPR v -> rows M=v (lanes 0-15) / M=v+8 (lanes 16-31), N = lane&15.
// ---------------------------------------------------------------------------
__global__ void __launch_bounds__(256)
node_proj_wmma_kernel(const float* __restrict__ nf,
                      const float* __restrict__ W,
                      float2* __restrict__ proj,
                      int n_nodes)
{
    const int lane = threadIdx.x & 31;
    const int waveInBlock = threadIdx.x >> 5;
    const int wavesPerBlock = blockDim.x >> 5;
    const int tile = blockIdx.x * wavesPerBlock + waveInBlock;
    const int nTiles = (n_nodes + 15) >> 4;
    if (tile >= nTiles) return;            // uniform per wave: EXEC stays full

    const int half = lane >> 4;            // 0 = lanes 0-15, 1 = lanes 16-31
    const int sub  = lane & 15;            // M index (for A) / N index (for B,D)

    int node = tile * 16 + sub;
    if (node > n_nodes - 1) node = n_nodes - 1;   // clamp (no divergence)
    const float* __restrict__ row = nf + (size_t)node * 128 + (half << 1);

    // Loop-invariant, branch-free B-column selection:
    //   lanes with sub==0 read column-0 weights (W[0:128]),
    //   lanes with sub==1 read column-1 weights (W[128:256]),
    //   lanes sub>=2 load harmlessly from column 0 and multiply by 0.
    const int   woff  = (sub == 1) ? 128 : 0;      // single v_cndmask
    const float wmask = (sub < 2) ? 1.0f : 0.0f;   // single v_cndmask
    const float* __restrict__ wrow = W + woff + (half << 1);

    v8f acc = {};
#pragma unroll 8
    for (int k0 = 0; k0 < 128; k0 += 4) {
        v2f a = *(const v2f*)(row + k0);    // global_load_b64 (A fragment)
        v2f w = *(const v2f*)(wrow + k0);   // global_load_b64 (weights, L0-hot)
        v2f b;
        b.x = w.x * wmask;                  // zero out lanes N>=2
        b.y = w.y * wmask;
        acc = __builtin_amdgcn_wmma_f32_16x16x4_f32(
            /*neg_a=*/false, a, /*neg_b=*/false, b,
            /*c_mod=*/(short)0, acc, /*reuse_a=*/false, /*reuse_b=*/false);
    }

    // Write out: p0 from lanes 0 (rows 0-7) / 16 (rows 8-15),
    //            p1 from lanes 1 (rows 0-7) / 17 (rows 8-15).
    if (sub < 2) {
        const int mbase = tile * 16 + half * 8;
#pragma unroll
        for (int v = 0; v < 8; ++v) {
            const int m = mbase + v;
            if (m < n_nodes) {
                float* p = reinterpret_cast<float*>(&proj[m]);
                p[sub] = acc[v];   // sub==0 -> p0, sub==1 -> p1
            }
        }
    }
}

// ---------------------------------------------------------------------------
// Kernel 2: per-edge gather + sigmoid.
// out[e] = sigmoid(proj[src].p0 + proj[dst].p1 + b)
// proj table is 800 KB -> lives in L2 (192 MB); random gathers stay on-chip.
// ---------------------------------------------------------------------------
__global__ void __launch_bounds__(256)
edge_sigmoid_kernel(const long long* __restrict__ ei,   // [2, E] int64
                    const float2* __restrict__ proj,
                    const float* __restrict__ bias,
                    float* __restrict__ out,
                    int n_edges)
{
    const int e = blockIdx.x * blockDim.x + threadIdx.x;
    if (e >= n_edges) return;
    const long long s = ei[e];                    // row 0: source nodes
    const long long d = ei[(size_t)n_edges + e];  // row 1: target nodes
    const float2 ps = proj[s];
    const float2 pd = proj[d];
    const float x = ps.x + pd.y + bias[0];
    out[e] = 1.0f / (1.0f + __expf(-x));
}

// ---------------------------------------------------------------------------
extern "C" void kernel_launch(void* const* d_in, const int* in_sizes, int n_in,
                              void* d_out, int out_size, void* d_ws, size_t ws_size,
                              hipStream_t stream)
{
    const float*     nf  = (const float*)d_in[0];      // [N, 128] f32
    const long long* ei  = (const long long*)d_in[1];  // [2, E] int64
    const float*     W   = (const float*)d_in[2];      // [256, 1] f32
    const float*     b   = (const float*)d_in[3];      // [1] f32

    const int n_nodes = in_sizes[0] / 128;
    const int n_edges = in_sizes[1] / 2;

    float2* proj = (float2*)d_ws;   // 8 bytes * n_nodes scratch

    // Kernel 1: 16-node tile per wave, 8 waves (256 threads) per block.
    const int nTiles = (n_nodes + 15) / 16;
    const int wavesPerBlock = 8;
    const int blocks1 = (nTiles + wavesPerBlock - 1) / wavesPerBlock;
    node_proj_wmma_kernel<<<blocks1, 256, 0, stream>>>(nf, W, proj, n_nodes);

    // Kernel 2: thread per edge.
    const int blocks2 = (n_edges + 255) / 256;
    edge_sigmoid_kernel<<<blocks2, 256, 0, stream>>>(ei, proj, b, (float*)d_out, n_edges);
}